// CommNet_90280212562554
// MI455X (gfx1250) — compile-verified
//
#include <hip/hip_runtime.h>

#define D 128
#define TILE 16
#define WAVES_PER_BLOCK 8

typedef __attribute__((ext_vector_type(2))) float v2f;
typedef __attribute__((ext_vector_type(8))) float v8f;

// Fused GEMM: out[m,n] = (accum? out : 0) + act( (A[m,:]*rowscale) @ W + bias[n] )
//   rowscale = 1/max(cnt[m],1) when cnt != nullptr (segment-mean fold-in)
// W (128x128 fp32, 64KB) is weight-stationary in LDS.
// Each wave computes a 16-row M-tile via V_WMMA_F32_16X16X4_F32, sweeping
// 8 N-tiles x 32 K-steps. A-panel (16x128) and per-tile B-panel (128x16) are
// register-resident; B is bulk-loaded so there is ONE DScnt wait per N-tile,
// and two independent accumulator chains give the matrix pipe ILP.
__global__ void __launch_bounds__(256)
gemm128_wmma(const float* __restrict__ A, const float* __restrict__ W,
             const float* __restrict__ bias, const float* __restrict__ cnt,
             float* __restrict__ out, int nrows, int do_relu, int do_accum)
{
    __shared__ float wlds[D * D];
    {   // cooperative 64KB weight stage (float4)
        const float4* s4 = (const float4*)W;
        float4* d4 = (float4*)wlds;
        for (int i = threadIdx.x; i < (D * D) / 4; i += blockDim.x) d4[i] = s4[i];
    }
    __syncthreads();

    const int lane = threadIdx.x & 31;
    const int wave = threadIdx.x >> 5;
    const int lo   = lane & 15;      // N / M-column index within tile
    const int hi   = lane >> 4;      // half-wave: selects K pair (ISA 7.12.2)
    const int tile = blockIdx.x * WAVES_PER_BLOCK + wave;
    const int ntiles = (nrows + TILE - 1) / TILE;
    if (tile >= ntiles) return;      // wave-uniform: EXEC all-ones at WMMA
    const int m0 = tile * TILE;

    int arow = m0 + lo;
    if (arow > nrows - 1) arow = nrows - 1;
    float scale = 1.0f;
    if (cnt) scale = 1.0f / fmaxf(cnt[arow], 1.0f);

    // Preload A-panel: per K-step s, lane holds cols (4s+2*hi, 4s+2*hi+1)
    // -> A-matrix 16x4 fp32 VGPR layout: V0 = K0|K2, V1 = K1|K3.
    v2f af[32];
    const float2* ap = (const float2*)(A + (size_t)arow * D);
    #pragma unroll
    for (int s = 0; s < 32; ++s) {
        float2 t = ap[2 * s + hi];
        af[s].x = t.x * scale;
        af[s].y = t.y * scale;
    }

    #pragma unroll 1
    for (int nt = 0; nt < D / TILE; ++nt) {
        const int n0 = nt * TILE;
        const float bval = bias[n0 + lo];

        // Bulk-load B panel (128x16) into registers: 32 clustered DS loads,
        // one DScnt wait, then an uninterrupted WMMA burst.
        v2f bf[32];
        #pragma unroll
        for (int s = 0; s < 32; ++s) {
            const int kb = 4 * s + 2 * hi;   // B 4x16: V0 = K0|K2, V1 = K1|K3
            bf[s].x = wlds[(kb    ) * D + n0 + lo];
            bf[s].y = wlds[(kb + 1) * D + n0 + lo];
        }

        v8f acc0 = {};
        v8f acc1 = {};
        #pragma unroll
        for (int s = 0; s < 32; s += 2) {
            acc0 = __builtin_amdgcn_wmma_f32_16x16x4_f32(
                false, af[s],     false, bf[s],     (short)0, acc0, false, false);
            acc1 = __builtin_amdgcn_wmma_f32_16x16x4_f32(
                false, af[s + 1], false, bf[s + 1], (short)0, acc1, false, false);
        }
        const v8f acc = acc0 + acc1;

        // C/D layout: VGPR j -> M = m0 + j + 8*hi, N = n0 + lo
        #pragma unroll
        for (int j = 0; j < 8; ++j) {
            const int m = m0 + j + 8 * hi;
            if (m < nrows) {
                float v = acc[j] + bval;
                if (do_relu) v = fmaxf(v, 0.0f);
                float* p = out + (size_t)m * D + n0 + lo;
                if (do_accum) v += *p;
                *p = v;
            }
        }
    }
}

// sums[src[e],:] += h[dst[e],:]; cnt[src[e]] += 1   (one wave per edge)
__global__ void __launch_bounds__(256)
edge_scatter(const float* __restrict__ h, const int* __restrict__ srcIdx,
             const int* __restrict__ dstIdx, float* __restrict__ sums,
             float* __restrict__ cnt, int nedges)
{
    const int e = blockIdx.x * 8 + (threadIdx.x >> 5);
    if (e >= nedges) return;
    const int lane = threadIdx.x & 31;
    const int s = srcIdx[e];
    const int d = dstIdx[e];
    const float4 v = *(const float4*)(h + (size_t)d * D + lane * 4);
    float* sp = sums + (size_t)s * D + lane * 4;
    atomicAdd(sp + 0, v.x);
    atomicAdd(sp + 1, v.y);
    atomicAdd(sp + 2, v.z);
    atomicAdd(sp + 3, v.w);
    if (lane == 0) atomicAdd(cnt + s, 1.0f);
}

__global__ void zero_f32(float* __restrict__ p, size_t n)
{
    size_t i = (size_t)blockIdx.x * blockDim.x + threadIdx.x;
    const size_t stride = (size_t)gridDim.x * blockDim.x;
    for (; i < n; i += stride) p[i] = 0.0f;
}

extern "C" void kernel_launch(void* const* d_in, const int* in_sizes, int n_in,
                              void* d_out, int out_size, void* d_ws, size_t ws_size,
                              hipStream_t stream)
{
    (void)n_in; (void)out_size; (void)ws_size;

    const float* x      = (const float*)d_in[0];
    const int*   edges  = (const int*)  d_in[1];  // [2,E] flat: src then dst
    const float* enc_w1 = (const float*)d_in[2];
    const float* enc_b1 = (const float*)d_in[3];
    const float* enc_w2 = (const float*)d_in[4];
    const float* enc_b2 = (const float*)d_in[5];
    const float* comm_w = (const float*)d_in[6];  // [R,128,128]
    const float* comm_b = (const float*)d_in[7];  // [R,128]
    const float* dec_w1 = (const float*)d_in[8];
    const float* dec_b1 = (const float*)d_in[9];
    const float* dec_w2 = (const float*)d_in[10];
    const float* dec_b2 = (const float*)d_in[11];

    const int N  = in_sizes[0] / D;
    const int E_ = in_sizes[1] / 2;
    const int R_ = in_sizes[6] / (D * D);

    float* h   = (float*)d_ws;                 // [N,128]
    float* tmp = h   + (size_t)N * D;          // [N,128] (also msg sums)
    float* cnt = tmp + (size_t)N * D;          // [N]

    const int ntiles = (N + TILE - 1) / TILE;
    const dim3 gblk(256), ggrd((ntiles + WAVES_PER_BLOCK - 1) / WAVES_PER_BLOCK);
    const int eblocks = (E_ + 7) / 8;

    // encoder
    gemm128_wmma<<<ggrd, gblk, 0, stream>>>(x,   enc_w1, enc_b1, nullptr, tmp, N, 1, 0);
    gemm128_wmma<<<ggrd, gblk, 0, stream>>>(tmp, enc_w2, enc_b2, nullptr, h,   N, 0, 0);

    // communication rounds: h += relu(mean_msg @ comm_w[r] + comm_b[r])
    for (int r = 0; r < R_; ++r) {
        zero_f32<<<2048, 256, 0, stream>>>(tmp, (size_t)N * D);
        zero_f32<<<512,  256, 0, stream>>>(cnt, (size_t)N);
        edge_scatter<<<eblocks, 256, 0, stream>>>(h, edges, edges + E_, tmp, cnt, E_);
        gemm128_wmma<<<ggrd, gblk, 0, stream>>>(tmp, comm_w + (size_t)r * D * D,
                                                comm_b + (size_t)r * D, cnt, h, N, 1, 1);
    }

    // decoder
    gemm128_wmma<<<ggrd, gblk, 0, stream>>>(h,   dec_w1, dec_b1, nullptr, tmp, N, 1, 0);
    gemm128_wmma<<<ggrd, gblk, 0, stream>>>(tmp, dec_w2, dec_b2, nullptr,
                                            (float*)d_out, N, 0, 0);
}